// customized_GAT_21388937134367
// MI455X (gfx1250) — compile-verified
//
#include <hip/hip_runtime.h>
#include <math.h>

#define N_NODES 100000
#define N_EDGES 1600000
#define IN_DIM 128
#define OUT_DIM 64
#define NEG_SLOPE 0.2f
// W is staged in LDS as float2 pairs along K: element (kp, n) = { W[2kp][n], W[2kp+1][n] }.
// Pair-row stride 65 (float2 units) => lanes 16-31 (kp+1) land on shifted banks vs lanes 0-15.
#define PAIR_STRIDE 65

typedef __attribute__((ext_vector_type(2))) float v2f;
typedef __attribute__((ext_vector_type(8))) float v8f;

__device__ __forceinline__ float lrelu(float x) {
  return x > 0.0f ? x : NEG_SLOPE * x;
}

// Float atomic max via sign-split int/uint atomics (monotone under IEEE bit order).
__device__ __forceinline__ void atomicMaxFloat(float* addr, float value) {
  if (value >= 0.0f)
    atomicMax((int*)addr, __float_as_int(value));
  else
    atomicMin((unsigned int*)addr, __float_as_uint(value));
}

// ---------------- Kernel 1: out = bias (broadcast), m = -inf, s = 0 ----------------
__global__ void gat_init_kernel(float* __restrict__ out, const float* __restrict__ bias,
                                float* __restrict__ m, float* __restrict__ s) {
  int tid = blockIdx.x * blockDim.x + threadIdx.x;
  if (tid < N_NODES * OUT_DIM) {
    int c = tid & (OUT_DIM - 1);
    out[tid] = bias[c];
    if (c == 0) {
      int i = tid >> 6;
      m[i] = -__builtin_inff();
      s[i] = 0.0f;
    }
  }
}

// ---------------- Kernel 2: h = leaky_relu(feats) @ W via V_WMMA_F32_16X16X4_F32,
//                  fused el = h@attn_l, er = h@attn_r ----------------
__global__ __launch_bounds__(256)
void gat_gemm_wmma_kernel(const float* __restrict__ feats, const float* __restrict__ W,
                          const float* __restrict__ attn_l, const float* __restrict__ attn_r,
                          float* __restrict__ h, float* __restrict__ el, float* __restrict__ er) {
  __shared__ v2f Wlds[(IN_DIM / 2) * PAIR_STRIDE];   // 64*65*8B = 33.3 KB of 320 KB
  const int tid = threadIdx.x;

  // Stage W [128x64] into LDS, interleaved along K so each B fragment is one b64 load.
  for (int idx = tid; idx < (IN_DIM / 2) * OUT_DIM; idx += 256) {
    int kp = idx >> 6, n = idx & 63;
    v2f p;
    p.x = W[(2 * kp) * OUT_DIM + n];
    p.y = W[(2 * kp + 1) * OUT_DIM + n];
    Wlds[kp * PAIR_STRIDE + n] = p;
  }
  __syncthreads();

  const int wave = tid >> 5;
  const int lane = tid & 31;
  const int tile = blockIdx.x * 8 + wave;       // 16-row tile of output
  if (tile * 16 >= N_NODES) return;             // wave-uniform; N divisible by 16

  const int half = lane >> 4;                   // 0: lanes 0-15, 1: lanes 16-31
  const int l16 = lane & 15;
  const int row = tile * 16 + l16;
  const float* __restrict__ arow = feats + (size_t)row * IN_DIM;

  v8f acc0 = {}; v8f acc1 = {}; v8f acc2 = {}; v8f acc3 = {};

  // ISA A layout (16x4 f32): lanes 0-15 hold K=k0,k0+1; lanes 16-31 hold K=k0+2,k0+3.
  #pragma unroll 4
  for (int k0 = 0; k0 < IN_DIM; k0 += 4) {
    const int kk = k0 + (half << 1);            // even -> 8B-aligned pair
    v2f a = *(const v2f*)(arow + kk);           // global_load_b64
    a.x = lrelu(a.x);
    a.y = lrelu(a.y);
    const v2f* wp = Wlds + ((k0 >> 1) + half) * PAIR_STRIDE + l16;
    v2f b0 = wp[0];                             // each: single ds_load_b64, even VGPR pair
    v2f b1 = wp[16];
    v2f b2 = wp[32];
    v2f b3 = wp[48];
    acc0 = __builtin_amdgcn_wmma_f32_16x16x4_f32(false, a, false, b0, (short)0, acc0, false, false);
    acc1 = __builtin_amdgcn_wmma_f32_16x16x4_f32(false, a, false, b1, (short)0, acc1, false, false);
    acc2 = __builtin_amdgcn_wmma_f32_16x16x4_f32(false, a, false, b2, (short)0, acc2, false, false);
    acc3 = __builtin_amdgcn_wmma_f32_16x16x4_f32(false, a, false, b3, (short)0, acc3, false, false);
  }

  // C/D layout: VGPR r -> (M = r + 8*half, N = l16).
  const int rowbase = tile * 16 + (half << 3);
  const float al0 = attn_l[l16], al1 = attn_l[l16 + 16], al2 = attn_l[l16 + 32], al3 = attn_l[l16 + 48];
  const float ar0 = attn_r[l16], ar1 = attn_r[l16 + 16], ar2 = attn_r[l16 + 32], ar3 = attn_r[l16 + 48];

  #pragma unroll
  for (int r = 0; r < 8; ++r) {
    float h0 = acc0[r], h1 = acc1[r], h2 = acc2[r], h3 = acc3[r];
    float* hp = h + (size_t)(rowbase + r) * OUT_DIM + l16;
    hp[0] = h0; hp[16] = h1; hp[32] = h2; hp[48] = h3;
    // Fused attention dots; reduce across the 16 lanes of each half (xor masks < 16).
    float pl = h0 * al0 + h1 * al1 + h2 * al2 + h3 * al3;
    float pr = h0 * ar0 + h1 * ar1 + h2 * ar2 + h3 * ar3;
    #pragma unroll
    for (int mm = 1; mm < 16; mm <<= 1) {
      pl += __shfl_xor(pl, mm, 32);
      pr += __shfl_xor(pr, mm, 32);
    }
    if (l16 == 0) {
      el[rowbase + r] = pl;
      er[rowbase + r] = pr;
    }
  }
}

// ---------------- Kernel 3: e = lrelu(el[src]+er[dst]); segment max into m ----------------
__global__ void gat_edge_logit_kernel(const int* __restrict__ src, const int* __restrict__ dst,
                                      const float* __restrict__ el, const float* __restrict__ er,
                                      float* __restrict__ e, float* __restrict__ m) {
  int j = blockIdx.x * blockDim.x + threadIdx.x;
  if (j < N_EDGES) {
    int dd = dst[j];
    float v = lrelu(el[src[j]] + er[dd]);
    e[j] = v;
    atomicMaxFloat(&m[dd], v);
  }
}

// ---------------- Kernel 4: e = exp(e - m[dst]); segment sum into s ----------------
__global__ void gat_edge_exp_kernel(const int* __restrict__ dst, float* __restrict__ e,
                                    const float* __restrict__ m, float* __restrict__ s) {
  int j = blockIdx.x * blockDim.x + threadIdx.x;
  if (j < N_EDGES) {
    int dd = dst[j];
    float v = __expf(e[j] - m[dd]);
    e[j] = v;
    atomicAdd(&s[dd], v);
  }
}

// ---------------- Kernel 5: out[dst] += (e/s[dst]) * h[src]  (64 threads / edge) ----------------
__global__ __launch_bounds__(256)
void gat_aggregate_kernel(const int* __restrict__ src, const int* __restrict__ dst,
                          const float* __restrict__ e, const float* __restrict__ s,
                          const float* __restrict__ h, float* __restrict__ out) {
  int t = blockIdx.x * 256 + threadIdx.x;
  int j = t >> 6;
  int c = t & 63;
  if (j < N_EDGES) {
    int sd = src[j], dd = dst[j];
    float a = e[j] / s[dd];                         // broadcast loads within the 64-thread group
    float val = a * h[(size_t)sd * OUT_DIM + c];    // gather: h is L2-resident (25.6 MB << 192 MB)
    atomicAdd(&out[(size_t)dd * OUT_DIM + c], val);
  }
}

extern "C" void kernel_launch(void* const* d_in, const int* in_sizes, int n_in,
                              void* d_out, int out_size, void* d_ws, size_t ws_size,
                              hipStream_t stream) {
  const float* feats  = (const float*)d_in[0];
  const float* W      = (const float*)d_in[1];
  const float* attn_l = (const float*)d_in[2];
  const float* attn_r = (const float*)d_in[3];
  const float* bias   = (const float*)d_in[4];
  const int*   src    = (const int*)d_in[5];
  const int*   dst    = (const int*)d_in[6];
  float* out = (float*)d_out;

  // Workspace layout (all fully rewritten each call -> replay-safe):
  float* h  = (float*)d_ws;                          // N*64
  float* el = h  + (size_t)N_NODES * OUT_DIM;        // N
  float* er = el + N_NODES;                          // N
  float* m  = er + N_NODES;                          // N
  float* s  = m  + N_NODES;                          // N
  float* e  = s  + N_NODES;                          // E

  gat_init_kernel<<<(N_NODES * OUT_DIM + 255) / 256, 256, 0, stream>>>(out, bias, m, s);

  const int ntiles = (N_NODES + 15) / 16;            // 6250
  gat_gemm_wmma_kernel<<<(ntiles + 7) / 8, 256, 0, stream>>>(feats, W, attn_l, attn_r, h, el, er);

  gat_edge_logit_kernel<<<(N_EDGES + 255) / 256, 256, 0, stream>>>(src, dst, el, er, e, m);
  gat_edge_exp_kernel<<<(N_EDGES + 255) / 256, 256, 0, stream>>>(dst, e, m, s);

  const int agg_blocks = (int)(((long long)N_EDGES * OUT_DIM + 255) / 256);  // 400000
  gat_aggregate_kernel<<<agg_blocks, 256, 0, stream>>>(src, dst, e, s, h, out);
}